// PPIConv_76373108457628
// MI455X (gfx1250) — compile-verified
//
#include <hip/hip_runtime.h>
#include <math.h>

#define NN     50000
#define INCH   256
#define HEADS  3
#define OUTCH  128
#define RR     3
#define EE     400000
#define SEMN   128
#define HC     384                 // HEADS*OUTCH
#define EPN    (EE + NN)           // edges + self loops
#define MT     (NN / 16)           // 3125 row tiles
#define NT_XP  (HC / 16)           // 24 col tiles
#define CT     4                   // col tiles per wave in xp GEMM
#define NG_XP  (NT_XP / CT)        // 6 col-tile groups
#define NT_SEM (SEMN / 16)         // 8 col tiles

typedef float v2f __attribute__((ext_vector_type(2)));
typedef float v8f __attribute__((ext_vector_type(8)));

// ---- order-preserving float <-> uint mapping for atomicMax on floats ----
__device__ __forceinline__ unsigned f2ord(float x) {
  unsigned u = __float_as_uint(x);
  return (u & 0x80000000u) ? ~u : (u | 0x80000000u);
}
__device__ __forceinline__ float ord2f(unsigned u) {
  unsigned b = (u & 0x80000000u) ? (u & 0x7FFFFFFFu) : ~u;
  return __uint_as_float(b);
}

// =========================== xp = x @ theta[r] ===========================
// One wave per 16x64 output strip (4 accumulators); A fragment loaded once
// per K-step as a per-lane-addressed float2 (K = k0 + 2*half), no selects.
__global__ void k_gemm_xp(const float* __restrict__ x,
                          const float* __restrict__ theta,
                          float* __restrict__ xp) {
  int wave = blockIdx.x * (blockDim.x >> 5) + (threadIdx.x >> 5);
  const int TOT = RR * MT * NG_XP;
  if (wave >= TOT) return;
  int r   = wave / (MT * NG_XP);
  int rem = wave - r * (MT * NG_XP);
  int mt  = rem / NG_XP;
  int ng  = rem - mt * NG_XP;
  int lane = threadIdx.x & 31;
  int lh = lane >> 4;            // K-split half of the wave
  int mr = lane & 15;            // row (A) / col (B,C) within tile
  const float* xrow = x + (size_t)(mt * 16 + mr) * INCH + 2 * lh;
  const float* th   = theta + (size_t)r * INCH * HC;
  int col0 = ng * (CT * 16) + mr;
  v8f c[CT] = {};
  for (int k0 = 0; k0 < INCH; k0 += 4) {
    v2f a = *(const v2f*)(xrow + k0);              // K = k0+2*lh, k0+2*lh+1
    const float* w0 = th + (size_t)(k0 + 2 * lh) * HC;
#pragma unroll
    for (int t = 0; t < CT; ++t) {
      int col = col0 + t * 16;
      v2f b;
      b[0] = w0[col];
      b[1] = w0[HC + col];
      c[t] = __builtin_amdgcn_wmma_f32_16x16x4_f32(false, a, false, b,
                                                   (short)0, c[t], false, false);
    }
  }
  float* xpr = xp + (size_t)r * NN * HC;
#pragma unroll
  for (int t = 0; t < CT; ++t) {
    int col = col0 + t * 16;
#pragma unroll
    for (int i = 0; i < 8; ++i)
      xpr[(size_t)(mt * 16 + i + 8 * lh) * HC + col] = c[t][i];
  }
}

// ================= a_s / a_d = per-head dots with att vecs ================
__global__ void k_attcoef(const float* __restrict__ xp,
                          const float* __restrict__ att_src,
                          const float* __restrict__ att_dst,
                          float* __restrict__ a_s, float* __restrict__ a_d) {
  int idx = blockIdx.x * blockDim.x + threadIdx.x;
  const int TOT = RR * NN * HEADS;
  if (idx >= TOT) return;
  int h = idx % HEADS;
  int n = (idx / HEADS) % NN;
  int r = idx / (NN * HEADS);
  const float* v  = xp + ((size_t)r * NN + n) * HC + h * OUTCH;
  const float* as = att_src + (r * HEADS + h) * OUTCH;
  const float* ad = att_dst + (r * HEADS + h) * OUTCH;
  float s0 = 0.f, s1 = 0.f;
  for (int c = 0; c < OUTCH; ++c) { s0 += v[c] * as[c]; s1 += v[c] * ad[c]; }
  a_s[idx] = s0;
  a_d[idx] = s1;
}

__device__ __forceinline__ void edge_nodes(const long long* __restrict__ ei,
                                           int r, int idx, int& src, int& dst) {
  if (idx < EE) {
    const long long* e = ei + (size_t)r * 2 * EE;
    src = (int)e[idx];
    dst = (int)e[EE + idx];
  } else {
    src = dst = idx - EE;        // self loop
  }
}

// =========================== edge softmax: max ===========================
__global__ void k_edge_max(const long long* __restrict__ ei,
                           const float* __restrict__ a_s,
                           const float* __restrict__ a_d,
                           unsigned* __restrict__ m_enc) {
  int gid = blockIdx.x * blockDim.x + threadIdx.x;
  if (gid >= RR * EPN) return;
  int r = gid / EPN, idx = gid - r * EPN;
  int src, dst;
  edge_nodes(ei, r, idx, src, dst);
  const float* as = a_s + ((size_t)r * NN + src) * HEADS;
  const float* ad = a_d + ((size_t)r * NN + dst) * HEADS;
  unsigned* mm = m_enc + ((size_t)r * NN + dst) * HEADS;
#pragma unroll
  for (int h = 0; h < HEADS; ++h) {
    float v = as[h] + ad[h];
    v = v > 0.f ? v : 0.2f * v;  // leaky_relu(0.2)
    atomicMax(&mm[h], f2ord(v));
  }
}

// =========================== edge softmax: sum ===========================
__global__ void k_edge_sum(const long long* __restrict__ ei,
                           const float* __restrict__ a_s,
                           const float* __restrict__ a_d,
                           const unsigned* __restrict__ m_enc,
                           float* __restrict__ z) {
  int gid = blockIdx.x * blockDim.x + threadIdx.x;
  if (gid >= RR * EPN) return;
  int r = gid / EPN, idx = gid - r * EPN;
  int src, dst;
  edge_nodes(ei, r, idx, src, dst);
  const float* as = a_s + ((size_t)r * NN + src) * HEADS;
  const float* ad = a_d + ((size_t)r * NN + dst) * HEADS;
  const unsigned* mm = m_enc + ((size_t)r * NN + dst) * HEADS;
  float* zz = z + ((size_t)r * NN + dst) * HEADS;
#pragma unroll
  for (int h = 0; h < HEADS; ++h) {
    float v = as[h] + ad[h];
    v = v > 0.f ? v : 0.2f * v;
    atomicAdd(&zz[h], expf(v - ord2f(mm[h])));
  }
}

// =================== aggregation: acc[dst] += xp[src]*alpha ==============
__global__ void k_edge_agg(const long long* __restrict__ ei,
                           const float* __restrict__ a_s,
                           const float* __restrict__ a_d,
                           const unsigned* __restrict__ m_enc,
                           const float* __restrict__ z,
                           const float* __restrict__ xp,
                           float* __restrict__ acc) {
  int bid = blockIdx.x;
  int r = bid / EPN, idx = bid - r * EPN;
  int src, dst;
  edge_nodes(ei, r, idx, src, dst);
  int c = threadIdx.x;           // 0..383
  int h = c >> 7;                // head = c / 128
  size_t sb = (size_t)r * NN + src;
  size_t db = (size_t)r * NN + dst;
  float v = a_s[sb * HEADS + h] + a_d[db * HEADS + h];
  v = v > 0.f ? v : 0.2f * v;
  float m = ord2f(m_enc[db * HEADS + h]);
  float alpha = expf(v - m) / (z[db * HEADS + h] + 1e-16f);
  atomicAdd(&acc[db * HC + c], xp[sb * HC + c] * alpha);
}

// ======================= bias + relu (in place on acc) ===================
__global__ void k_bias_relu(float* __restrict__ acc,
                            const float* __restrict__ bias) {
  size_t i = (size_t)blockIdx.x * blockDim.x + threadIdx.x;
  if (i >= (size_t)RR * NN * HC) return;
  int c = (int)(i % HC);
  int r = (int)(i / ((size_t)NN * HC));
  acc[i] = fmaxf(acc[i] + bias[r * HC + c], 0.f);
}

// ====== semantic attention scores: score[r][n] = sum_s tanh(.)*q_sem =====
// One wave computes 16 rows x 128 cols (8 WMMA col-tiles), K = 384.
__global__ void k_sem_score(const float* __restrict__ acc,
                            const float* __restrict__ Wsem,
                            const float* __restrict__ b_sem,
                            const float* __restrict__ q_sem,
                            float* __restrict__ score) {
  int wave = blockIdx.x * (blockDim.x >> 5) + (threadIdx.x >> 5);
  const int TOT = RR * MT;
  if (wave >= TOT) return;
  int r = wave / MT, mt = wave - r * MT;
  int lane = threadIdx.x & 31;
  int lh = lane >> 4, mr = lane & 15;
  const float* arow =
      acc + ((size_t)r * NN + mt * 16 + mr) * HC + 2 * lh;
  v8f c[NT_SEM] = {};
  for (int k0 = 0; k0 < HC; k0 += 4) {
    v2f a = *(const v2f*)(arow + k0);              // K = k0+2*lh, k0+2*lh+1
    const float* w0 = Wsem + (size_t)(k0 + 2 * lh) * SEMN;
#pragma unroll
    for (int t = 0; t < NT_SEM; ++t) {
      int colt = t * 16 + mr;
      v2f b;
      b[0] = w0[colt];
      b[1] = w0[SEMN + colt];
      c[t] = __builtin_amdgcn_wmma_f32_16x16x4_f32(false, a, false, b,
                                                   (short)0, c[t], false, false);
    }
  }
  float partial[8];
#pragma unroll
  for (int i = 0; i < 8; ++i) partial[i] = 0.f;
#pragma unroll
  for (int t = 0; t < NT_SEM; ++t) {
    int colt = t * 16 + mr;
    float bb = b_sem[colt], qq = q_sem[colt];
#pragma unroll
    for (int i = 0; i < 8; ++i)
      partial[i] += tanhf(c[t][i] + bb) * qq;
  }
  // reduce over the 16 lanes holding the same rows
#pragma unroll
  for (int i = 0; i < 8; ++i)
    for (int off = 8; off > 0; off >>= 1)
      partial[i] += __shfl_xor(partial[i], off, 16);
  if (mr == 0) {
#pragma unroll
    for (int i = 0; i < 8; ++i)
      score[(size_t)r * NN + mt * 16 + i + 8 * lh] = partial[i];
  }
}

// ============ final: beta softmax over R + metagraph residual ============
__global__ void k_final(const float* __restrict__ acc,
                        const float* __restrict__ score,
                        const float* __restrict__ mgx,
                        const float* __restrict__ gsrc,
                        const float* __restrict__ gdst,
                        float* __restrict__ out) {
  size_t i = (size_t)blockIdx.x * blockDim.x + threadIdx.x;
  if (i >= (size_t)NN * HC) return;
  int n = (int)(i / HC);
  int c = (int)(i % HC);
  float s0 = score[n], s1 = score[NN + n], s2 = score[2 * NN + n];
  float m = fmaxf(s0, fmaxf(s1, s2));
  float e0 = expf(s0 - m), e1 = expf(s1 - m), e2 = expf(s2 - m);
  float inv = 1.f / (e0 + e1 + e2);
  float v = e0 * inv * acc[i] +
            e1 * inv * acc[(size_t)NN * HC + i] +
            e2 * inv * acc[2 * (size_t)NN * HC + i];
  out[i] = v + mgx[c] * (gsrc[c] + gdst[c]);
}

// ================================ zero ===================================
__global__ void k_zero(unsigned* __restrict__ p, size_t n) {
  size_t i = (size_t)blockIdx.x * blockDim.x + threadIdx.x;
  size_t stride = (size_t)gridDim.x * blockDim.x;
  for (; i < n; i += stride) p[i] = 0u;
}

extern "C" void kernel_launch(void* const* d_in, const int* in_sizes, int n_in,
                              void* d_out, int out_size, void* d_ws, size_t ws_size,
                              hipStream_t stream) {
  const float*     x        = (const float*)d_in[0];
  const float*     theta    = (const float*)d_in[1];
  const float*     att_src  = (const float*)d_in[2];
  const float*     att_dst  = (const float*)d_in[3];
  const float*     gat_bias = (const float*)d_in[4];
  const float*     W_sem    = (const float*)d_in[5];
  const float*     b_sem    = (const float*)d_in[6];
  const float*     q_sem    = (const float*)d_in[7];
  const float*     mgx      = (const float*)d_in[8];
  const float*     gsrc     = (const float*)d_in[9];
  const float*     gdst     = (const float*)d_in[10];
  const long long* ei       = (const long long*)d_in[11];
  float* out = (float*)d_out;

  // ---- carve workspace ----
  char* ws = (char*)d_ws;
  size_t off = 0;
  auto carve = [&](size_t bytes) -> char* {
    off = (off + 255) & ~(size_t)255;
    char* p = ws + off;
    off += bytes;
    return p;
  };
  float*    xp    = (float*)carve((size_t)RR * NN * HC * 4);
  float*    acc   = (float*)carve((size_t)RR * NN * HC * 4);
  float*    a_s   = (float*)carve((size_t)RR * NN * HEADS * 4);
  float*    a_d   = (float*)carve((size_t)RR * NN * HEADS * 4);
  unsigned* m_enc = (unsigned*)carve((size_t)RR * NN * HEADS * 4);
  float*    z     = (float*)carve((size_t)RR * NN * HEADS * 4);
  float*    score = (float*)carve((size_t)RR * NN * 4);
  (void)ws_size; (void)in_sizes; (void)n_in; (void)out_size;

  // ---- init accumulators ----
  k_zero<<<4096, 256, 0, stream>>>((unsigned*)acc, (size_t)RR * NN * HC);
  k_zero<<<512, 256, 0, stream>>>(m_enc, (size_t)RR * NN * HEADS);
  k_zero<<<512, 256, 0, stream>>>((unsigned*)z, (size_t)RR * NN * HEADS);

  // ---- xp = x @ theta[r]  (WMMA f32, 16x64 strips) ----
  {
    int waves = RR * MT * NG_XP;              // 56250
    k_gemm_xp<<<(waves + 3) / 4, 128, 0, stream>>>(x, theta, xp);
  }

  // ---- attention coefficients ----
  k_attcoef<<<(RR * NN * HEADS + 255) / 256, 256, 0, stream>>>(
      xp, att_src, att_dst, a_s, a_d);

  // ---- edge softmax + aggregation ----
  k_edge_max<<<(RR * EPN + 255) / 256, 256, 0, stream>>>(ei, a_s, a_d, m_enc);
  k_edge_sum<<<(RR * EPN + 255) / 256, 256, 0, stream>>>(ei, a_s, a_d, m_enc, z);
  k_edge_agg<<<RR * EPN, HC, 0, stream>>>(ei, a_s, a_d, m_enc, z, xp, acc);

  // ---- bias + relu ----
  {
    size_t tot = (size_t)RR * NN * HC;
    k_bias_relu<<<(unsigned)((tot + 255) / 256), 256, 0, stream>>>(acc, gat_bias);
  }

  // ---- semantic attention scores (WMMA f32, 16x128 strips) ----
  {
    int waves = RR * MT;                      // 9375
    k_sem_score<<<(waves + 3) / 4, 128, 0, stream>>>(acc, W_sem, b_sem, q_sem, score);
  }

  // ---- final mix ----
  {
    size_t tot = (size_t)NN * HC;
    k_final<<<(unsigned)((tot + 255) / 256), 256, 0, stream>>>(
        acc, score, mgx, gsrc, gdst, out);
  }
}